// RegionLoss_84490596647415
// MI455X (gfx1250) — compile-verified
//
#include <hip/hip_runtime.h>
#include <math.h>

typedef float v2f __attribute__((ext_vector_type(2)));
typedef float v8f __attribute__((ext_vector_type(8)));

#define NB 64
#define NA 5
#define NC 80
#define NH 38
#define NW 38
#define PLANE (NH * NW)                    /* 1444 */
#define NPLANES (NB * NA)                  /* 320  */
#define GROUPS_PER_PLANE ((PLANE + 15)/16) /* 91   */
#define NCELLS (NPLANES * PLANE)           /* 462080 */

#if defined(__has_builtin)
#if __has_builtin(__builtin_amdgcn_wmma_f32_16x16x4_f32)
#define USE_WMMA_F32X4 1
#endif
#endif

// ws layout: [0..11] accumulators (sumP f32, sumM f32, maxBits i32), [256..] mask bytes
__global__ void rl_init_kernel(unsigned int* mask_w, float* acc) {
    int i = blockIdx.x * blockDim.x + threadIdx.x;
    if (i < NCELLS / 4) mask_w[i] = 0u;           // zero mask bytes (NCELLS % 4 == 0)
    if (i == 0) {
        acc[0] = 0.0f;                            // sum(p*m)
        acc[1] = 0.0f;                            // sum(m)
        ((int*)acc)[2] = (int)0xff800000;         // -inf bits for masked max
    }
}

__global__ void rl_mask_kernel(const float* __restrict__ target,
                               unsigned char* __restrict__ mask) {
    const float AW[5] = {0.57273f, 1.87446f, 3.33843f, 7.88282f, 9.77052f};
    const float AH[5] = {0.677385f, 2.06253f, 5.47434f, 3.52778f, 9.16828f};
    int b = threadIdx.x;
    if (b >= NB) return;
    const float* t = target + b * 250;
    for (int i = 0; i < 50; ++i) {
        float f1 = t[i * 5 + 1];
        if (f1 == 0.0f) break;                    // cumprod validity gate
        float gx = f1 * (float)NW;
        float gy = t[i * 5 + 2] * (float)NH;
        float gw = t[i * 5 + 3] * (float)NW;
        float gh = t[i * 5 + 4] * (float)NH;
        int gi = (int)floorf(gx); gi = gi < 0 ? 0 : (gi > NW - 1 ? NW - 1 : gi);
        int gj = (int)floorf(gy); gj = gj < 0 ? 0 : (gj > NH - 1 ? NH - 1 : gj);
        float best = -1.0f; int bn = 0;
        #pragma unroll
        for (int a = 0; a < NA; ++a) {
            float inter = fminf(gw, AW[a]) * fminf(gh, AH[a]);
            float iou = inter / (gw * gh + AW[a] * AH[a] - inter);
            if (iou > best) { best = iou; bn = a; } // first-max wins
        }
        int flat = ((b * NA + bn) * NH + gj) * NW + gi;
        mask[flat] = 1;
    }
}

__global__ __launch_bounds__(128)
void rl_softmax_reduce_kernel(const float* __restrict__ in,
                              const unsigned char* __restrict__ mask,
                              const int* __restrict__ reqd,
                              float* __restrict__ acc) {
    const int tid  = threadIdx.x;
    const int lane = tid & 31;
    const int wid  = blockIdx.x * (blockDim.x >> 5) + (tid >> 5);
    const int plane = wid / GROUPS_PER_PLANE;
    const int grp   = wid % GROUPS_PER_PLANE;
    const int b = plane / NA;
    const int a = plane % NA;
    const int M    = lane & 15;   // cell (A-matrix row) this lane serves
    const int half = lane >> 4;   // 0: classes 4t,4t+1 ; 1: classes 4t+2,4t+3
    const int cell  = grp * 16 + M;
    const int cellc = cell < PLANE ? cell : PLANE - 1;     // clamp tail group
    const size_t base = (size_t)(b * 425 + a * 85 + 5) * PLANE + (size_t)cellc;

    // Cache this cell's 40 assigned class logits (hardware 16x4-f32 A layout).
    float ea[40];
    #pragma unroll
    for (int t = 0; t < 20; ++t) {
        int cls = 4 * t + 2 * half;
        ea[2 * t]     = in[base + (size_t)cls * PLANE];
        ea[2 * t + 1] = in[base + (size_t)(cls + 1) * PLANE];
    }

    // Per-cell max across both lane halves.
    float mx = ea[0];
    #pragma unroll
    for (int j = 1; j < 40; ++j) mx = fmaxf(mx, ea[j]);
    mx = fmaxf(mx, __shfl_xor(mx, 16));

    #pragma unroll
    for (int j = 0; j < 40; ++j) ea[j] = __expf(ea[j] - mx);

    // Sum over 80 classes per cell: D = A(16x4 exps) * ones(4x16) + C, 20 K-chunks.
#ifdef USE_WMMA_F32X4
    v8f c8 = {};
    v2f ones; ones.x = 1.0f; ones.y = 1.0f;
    #pragma unroll
    for (int t = 0; t < 20; ++t) {
        v2f A; A.x = ea[2 * t]; A.y = ea[2 * t + 1];
        c8 = __builtin_amdgcn_wmma_f32_16x16x4_f32(false, A, false, ones,
                                                   (short)0, c8, false, false);
    }
    // Row M's sum: VGPR M on lanes 0-15 (M<8), VGPR M-8 on lanes 16-31 (M>=8).
    float sw[8];
    #pragma unroll
    for (int j = 0; j < 8; ++j) sw[j] = __shfl_xor(c8[j], 16);
    int idx = M & 7;
    float own = c8[0], oth = sw[0];
    #pragma unroll
    for (int j = 1; j < 8; ++j) { if (idx == j) { own = c8[j]; oth = sw[j]; } }
    float sum = (M < 8) ? oth : own;
#else
    float s = 0.0f;
    #pragma unroll
    for (int j = 0; j < 40; ++j) s += ea[j];
    float sum = s + __shfl_xor(s, 16);
#endif

    // Requested-class prob + masked reduction (lanes 16-31 own one cell each).
    const int creq = reqd[0];
    float val = 0.0f, cnt = 0.0f;
    int mb = (int)0xff800000;
    if (half == 1 && cell < PLANE) {
        float lc = in[base + (size_t)creq * PLANE];
        float p = __expf(lc - mx) / sum;
        int flat = plane * PLANE + cell;
        if (mask[flat]) { val = p; cnt = 1.0f; mb = __float_as_int(p); }
    }
    #pragma unroll
    for (int o = 16; o > 0; o >>= 1) {
        val += __shfl_xor(val, o);
        cnt += __shfl_xor(cnt, o);
        int ob = __shfl_xor(mb, o);
        mb = ob > mb ? ob : mb;
    }
    if (lane == 0) {
        atomicAdd(&acc[0], val);
        atomicAdd(&acc[1], cnt);
        atomicMax((int*)acc + 2, mb);
    }
}

__global__ void rl_finalize_kernel(const float* __restrict__ acc,
                                   float* __restrict__ out) {
    if (threadIdx.x == 0 && blockIdx.x == 0) {
        out[0] = acc[0] / acc[1];
        out[1] = __int_as_float(((const int*)acc)[2]);
    }
}

extern "C" void kernel_launch(void* const* d_in, const int* in_sizes, int n_in,
                              void* d_out, int out_size, void* d_ws, size_t ws_size,
                              hipStream_t stream) {
    (void)in_sizes; (void)n_in; (void)out_size; (void)ws_size;
    const float* output = (const float*)d_in[0];
    const float* target = (const float*)d_in[1];
    const int*   reqd   = (const int*)d_in[2];
    float* out = (float*)d_out;

    float* acc = (float*)d_ws;
    unsigned char* mask = (unsigned char*)d_ws + 256;

    rl_init_kernel<<<(NCELLS / 4 + 255) / 256, 256, 0, stream>>>((unsigned int*)mask, acc);
    rl_mask_kernel<<<1, 64, 0, stream>>>(target, mask);

    const int total_waves = NPLANES * GROUPS_PER_PLANE;  // 29120, divisible by 4
    rl_softmax_reduce_kernel<<<total_waves / 4, 128, 0, stream>>>(output, mask, reqd, acc);
    rl_finalize_kernel<<<1, 1, 0, stream>>>(acc, out);
}